// CtcDecodeLayer_70153995813306
// MI455X (gfx1250) — compile-verified
//
#include <hip/hip_runtime.h>

// Problem constants (match reference)
constexpr int kB     = 64;
constexpr int kT     = 256;
constexpr int kC     = 96;
constexpr int kBeam  = 100;
constexpr int kBlank = kC - 1;        // 95
#define NEGV (-1.0e30f)

constexpr int kThreads = 256;                          // 8 wave32s
constexpr int kNE    = kBeam * kC;                     // 9600 candidate scores
constexpr int kEPT   = (kNE + kThreads - 1) / kThreads;// 38 elements per thread
constexpr int kWords = (kNE + 31) / 32;                // 300 bitmask words

__device__ __forceinline__ float logaddexp_f(float a, float b) {
    float mx = fmaxf(a, b);
    float mn = fminf(a, b);
    return mx + log1pf(expf(mn - mx));
}

// Monotonic float->uint key: descending float order == descending key order.
__device__ __forceinline__ unsigned key_of(float f) {
    unsigned u = __float_as_uint(f);
    return (u & 0x80000000u) ? ~u : (u | 0x80000000u);
}
__device__ __forceinline__ float val_of(unsigned k) {
    unsigned u = (k & 0x80000000u) ? (k ^ 0x80000000u) : ~k;
    return __uint_as_float(u);
}

__global__ __launch_bounds__(kThreads)
void ctc_beam_kernel(const float* __restrict__ x,        // [B, T, C] post-softmax probs
                     const int*   __restrict__ lens,     // [B]
                     float*       __restrict__ out,      // [B, T] decoded labels (as float)
                     unsigned short* __restrict__ bp_ws) // [B, T, BEAM] packed backpointers
{
    // ---- LDS state ----
    __shared__ float lpb_s[kBeam], lpnb_s[kBeam], lse_s[kBeam];
    __shared__ float stayb_s[kBeam], staynb_s[kBeam], staytot_s[kBeam];
    __shared__ int   last_s[kBeam], plen_s[kBeam];
    __shared__ float nlpb_s[kBeam], nlpnb_s[kBeam];
    __shared__ int   nlast_s[kBeam], nplen_s[kBeam];
    __shared__ __align__(16) float raw_s[2][kC];   // double-buffered async landing zone
    __shared__ float lp_s[kC];                     // log(x + 1e-7) for current t
    // top-k machinery
    __shared__ int      hist_s[256];
    __shared__ int      wtot_s[8];
    __shared__ unsigned pref_bc_s;
    __shared__ int      kneed_bc_s;
    __shared__ unsigned bm_s[kWords];              // equal-to-threshold bitmask
    __shared__ unsigned gtk_s[kBeam];              // keys  > threshold (unordered)
    __shared__ int      gti_s[kBeam];              // idxs  > threshold (unordered)
    __shared__ int      cntgt_s;
    __shared__ unsigned sel_key_s[kBeam];          // rank-ordered selection
    __shared__ int      sel_idx_s[kBeam];
    // reconstruction
    __shared__ int seq_s[kT];
    __shared__ int bplen_s;

    const int tid = threadIdx.x;
    const int b   = blockIdx.x;
    const int lane = tid & 31;
    const int wid  = tid >> 5;
    const int seqlen = lens[b];
    const float* xb = x + (size_t)b * kT * kC;

    for (int i = tid; i < kBeam; i += kThreads) {
        lpb_s[i]  = (i == 0) ? 0.0f : NEGV;
        lpnb_s[i] = NEGV;
        last_s[i] = -1;
        plen_s[i] = 0;
    }

    // ---- async prefetch of row t=0 into raw_s[0] (24 lanes x b128 = 96 floats) ----
    if (tid < kC / 4) {
        unsigned lds0 = (unsigned)(uintptr_t)(&raw_s[0][0]) + (unsigned)tid * 16u;
        unsigned long long ga = (unsigned long long)(uintptr_t)xb + (unsigned long long)tid * 16ull;
        asm volatile("global_load_async_to_lds_b128 %0, %1, off"
                     :: "v"(lds0), "v"(ga) : "memory");
    }
    __syncthreads();

    for (int t = 0; t < kT; ++t) {
        const int cur = t & 1;

        // Prefetch next row; asynccnt<=1 guarantees the *older* load (row t)
        // has landed while row t+1 is still in flight (async loads are in-order).
        if (t + 1 < kT) {
            if (tid < kC / 4) {
                unsigned lds1 = (unsigned)(uintptr_t)(&raw_s[cur ^ 1][0]) + (unsigned)tid * 16u;
                unsigned long long ga = (unsigned long long)(uintptr_t)(xb + (size_t)(t + 1) * kC)
                                        + (unsigned long long)tid * 16ull;
                asm volatile("global_load_async_to_lds_b128 %0, %1, off"
                             :: "v"(lds1), "v"(ga) : "memory");
            }
            asm volatile("s_wait_asynccnt 0x1" ::: "memory");
        } else {
            asm volatile("s_wait_asynccnt 0x0" ::: "memory");
        }
        __syncthreads();

        if (tid < kC) lp_s[tid] = logf(raw_s[cur][tid] + 1e-7f);
        const bool active = (t < seqlen);   // uniform across block
        __syncthreads();

        if (active) {
            // ---- per-beam 'stay' quantities ----
            if (tid < kBeam) {
                float a  = lpb_s[tid], nb = lpnb_s[tid];
                float lse = logaddexp_f(a, nb);
                lse_s[tid] = lse;
                float sb = lse + lp_s[kBlank];
                int   l  = last_s[tid];
                float snb = (l >= 0) ? (nb + lp_s[l]) : NEGV;
                stayb_s[tid]   = sb;
                staynb_s[tid]  = snb;
                staytot_s[tid] = logaddexp_f(sb, snb);
            }
            __syncthreads();

            // ---- per-thread register-resident score keys ----
            unsigned kk[kEPT];
            #pragma unroll
            for (int j = 0; j < kEPT; ++j) {
                int e = tid + kThreads * j;
                unsigned key = 0u;                 // padding: smaller than any real key
                if (e < kNE) {
                    int i = e / kC;
                    int c = e - i * kC;
                    float val;
                    if (c == kBlank) {
                        val = staytot_s[i];
                    } else {
                        float base = (c == last_s[i]) ? lpb_s[i] : lse_s[i];
                        val = base + lp_s[c];
                    }
                    key = key_of(val);
                }
                kk[j] = key;
            }

            // ---- radix-select the 100th-largest key (MSB-first, 4x8-bit) ----
            unsigned prefix = 0, maskhi = 0;
            int kneed = kBeam;
            for (int shift = 24; shift >= 0; shift -= 8) {
                hist_s[tid] = 0;
                __syncthreads();
                #pragma unroll
                for (int j = 0; j < kEPT; ++j) {
                    int e = tid + kThreads * j;
                    if (e < kNE && ((kk[j] ^ prefix) & maskhi) == 0)
                        atomicAdd(&hist_s[(kk[j] >> shift) & 255], 1);
                }
                __syncthreads();
                // descending-digit inclusive prefix sum (wave32 shuffle scan)
                int digit = 255 - tid;
                int cnt = hist_s[digit];
                int s = cnt;
                #pragma unroll
                for (int off = 1; off < 32; off <<= 1) {
                    int n = __shfl_up(s, off, 32);
                    if (lane >= off) s += n;
                }
                if (lane == 31) wtot_s[wid] = s;
                __syncthreads();
                int woff = 0;
                #pragma unroll
                for (int w = 0; w < 8; ++w) if (w < wid) woff += wtot_s[w];
                int cum   = s + woff;    // count of matching keys with digit >= `digit`
                int cumex = cum - cnt;
                if (cumex < kneed && kneed <= cum) {   // unique crossing thread
                    pref_bc_s  = prefix | ((unsigned)digit << shift);
                    kneed_bc_s = kneed - cumex;
                }
                __syncthreads();
                prefix = pref_bc_s;
                kneed  = kneed_bc_s;
                maskhi |= (0xFFu << shift);
            }
            const unsigned thresh = prefix;
            const int count_gt = kBeam - kneed;   // keys strictly above threshold

            // ---- collect >threshold set and ==threshold bitmask ----
            if (tid == 0) cntgt_s = 0;
            for (int w = tid; w < kWords; w += kThreads) bm_s[w] = 0u;
            __syncthreads();
            #pragma unroll
            for (int j = 0; j < kEPT; ++j) {
                int e = tid + kThreads * j;
                if (e < kNE) {
                    if (kk[j] > thresh) {
                        int p = atomicAdd(&cntgt_s, 1);
                        gtk_s[p] = kk[j];
                        gti_s[p] = e;
                    } else if (kk[j] == thresh) {
                        atomicOr(&bm_s[e >> 5], 1u << (e & 31));
                    }
                }
            }
            __syncthreads();

            // rank the >threshold elements: (key desc, index asc) — O(n^2), n<=99
            const int ng = cntgt_s;
            if (tid < ng) {
                unsigned mk = gtk_s[tid];
                int      mi = gti_s[tid];
                int rank = 0;
                for (int i = 0; i < ng; ++i) {
                    unsigned ok = gtk_s[i];
                    int      oi = gti_s[i];
                    rank += (ok > mk || (ok == mk && oi < mi)) ? 1 : 0;
                }
                sel_key_s[rank] = mk;
                sel_idx_s[rank] = mi;
            }

            // take first `kneed` ==threshold elements in ascending flat index
            int c0 = 0, c1 = 0;
            if (tid < 150) {
                c0 = __popc(bm_s[2 * tid]);
                if (2 * tid + 1 < kWords) c1 = __popc(bm_s[2 * tid + 1]);
            }
            int sum = c0 + c1;
            int ss = sum;
            #pragma unroll
            for (int off = 1; off < 32; off <<= 1) {
                int n = __shfl_up(ss, off, 32);
                if (lane >= off) ss += n;
            }
            if (lane == 31) wtot_s[wid] = ss;
            __syncthreads();
            int woff = 0;
            #pragma unroll
            for (int w = 0; w < 8; ++w) if (w < wid) woff += wtot_s[w];
            int excl = ss + woff - sum;   // eq-elements before this thread's word pair
            if (tid < 150) {
                int rr = excl;
                unsigned wv0 = bm_s[2 * tid];
                while (wv0 && rr < kneed) {
                    int bit = __ffs(wv0) - 1;
                    sel_key_s[count_gt + rr] = thresh;
                    sel_idx_s[count_gt + rr] = (2 * tid) * 32 + bit;
                    ++rr;
                    wv0 &= wv0 - 1;
                }
                if (2 * tid + 1 < kWords) {
                    rr = excl + c0;
                    unsigned wv1 = bm_s[2 * tid + 1];
                    while (wv1 && rr < kneed) {
                        int bit = __ffs(wv1) - 1;
                        sel_key_s[count_gt + rr] = thresh;
                        sel_idx_s[count_gt + rr] = (2 * tid + 1) * 32 + bit;
                        ++rr;
                        wv1 &= wv1 - 1;
                    }
                }
            }
            __syncthreads();

            // ---- build new beams (fully parallel, 100 threads) ----
            if (tid < kBeam) {
                int r = tid;
                int idx = sel_idx_s[r];
                int bi = idx / kC;
                int ch = idx - bi * kC;
                int is_stay = (ch == kBlank) ? 1 : 0;
                float wv = val_of(sel_key_s[r]);
                nlpb_s[r]  = is_stay ? stayb_s[bi]  : NEGV;
                nlpnb_s[r] = is_stay ? staynb_s[bi] : wv;
                nlast_s[r] = is_stay ? last_s[bi]   : ch;
                nplen_s[r] = plen_s[bi] + (is_stay ? 0 : 1);
                bp_ws[((size_t)b * kT + t) * kBeam + r] =
                    (unsigned short)(bi | (ch << 7) | (is_stay << 14));
            }
            __syncthreads();
            if (tid < kBeam) {
                lpb_s[tid]  = nlpb_s[tid];
                lpnb_s[tid] = nlpnb_s[tid];
                last_s[tid] = nlast_s[tid];
                plen_s[tid] = nplen_s[tid];
            }
            __syncthreads();
        } else {
            // inactive timestep: identity backpointers so backtracking is uniform
            if (tid < kBeam)
                bp_ws[((size_t)b * kT + t) * kBeam + tid] =
                    (unsigned short)(tid | (1 << 14));
            __syncthreads();
        }
    }

    // ---- pick best beam (argmax = first max, like jnp.argmax) and backtrack ----
    if (tid == 0) {
        float bvv = -INFINITY; int bbi = 0;
        for (int i = 0; i < kBeam; ++i) {
            float tot = logaddexp_f(lpb_s[i], lpnb_s[i]);
            if (tot > bvv) { bvv = tot; bbi = i; }
        }
        bplen_s = plen_s[bbi];
        int cur = bbi;
        int p = plen_s[bbi] - 1;
        for (int t2 = kT - 1; t2 >= 0; --t2) {
            int bp = bp_ws[((size_t)b * kT + t2) * kBeam + cur];
            int stay = (bp >> 14) & 1;
            if (!stay) { seq_s[p] = (bp >> 7) & 127; --p; }
            cur = bp & 127;
        }
    }
    __syncthreads();
    int plenb = bplen_s;
    out[(size_t)b * kT + tid] = (tid < plenb) ? (float)seq_s[tid] : -1.0f;
}

extern "C" void kernel_launch(void* const* d_in, const int* in_sizes, int n_in,
                              void* d_out, int out_size, void* d_ws, size_t ws_size,
                              hipStream_t stream) {
    const float* x    = (const float*)d_in[0];   // [64, 256, 96] float32
    const int*   lens = (const int*)d_in[1];     // [64, 1] int32
    float* out = (float*)d_out;                  // [64, 256]
    unsigned short* bp = (unsigned short*)d_ws;  // needs 64*256*100*2 = 3.3 MB

    hipLaunchKernelGGL(ctc_beam_kernel, dim3(kB), dim3(kThreads), 0, stream,
                       x, lens, out, bp);
}